// StructureAwareLoss_16733192585747
// MI455X (gfx1250) — compile-verified
//
#include <hip/hip_runtime.h>
#include <hip/hip_bf16.h>
#include <math.h>
#include <stdint.h>

// ---- problem constants (match reference) ----
#define VOCAB      32000
#define BATCH      4
#define SEQ        1024
#define ROWS       (BATCH * SEQ)     // 4096
#define NCHUNK     4
#define CHUNK      (VOCAB / NCHUNK)  // 8000 floats = 32 KB LDS
#define CHUNK_V4   (CHUNK / 4)       // 2000 float4
#define TPB        256               // 8 waves (wave32)

#define PAD_IDX      0
#define NEWLINE_IDX  1
#define EOS_IDX      2

// ---------------- block reductions (wave32-aware) ----------------
__device__ __forceinline__ float warpMax32(float v) {
#pragma unroll
  for (int o = 16; o > 0; o >>= 1) v = fmaxf(v, __shfl_xor(v, o, 32));
  return v;
}
__device__ __forceinline__ float warpSum32(float v) {
#pragma unroll
  for (int o = 16; o > 0; o >>= 1) v += __shfl_xor(v, o, 32);
  return v;
}
// TPB == 256 -> 8 waves. scr must hold >= 8 floats.
__device__ __forceinline__ float blockMax(float v, float* scr) {
  v = warpMax32(v);
  const int w = threadIdx.x >> 5, l = threadIdx.x & 31;
  if (l == 0) scr[w] = v;
  __syncthreads();
  float r = (l < (TPB >> 5)) ? scr[l] : -__builtin_inff();
  r = warpMax32(r);          // every wave reduces the 8 partials itself
  __syncthreads();           // scr reusable after return
  return r;
}
__device__ __forceinline__ float blockSum(float v, float* scr) {
  v = warpSum32(v);
  const int w = threadIdx.x >> 5, l = threadIdx.x & 31;
  if (l == 0) scr[w] = v;
  __syncthreads();
  float r = (l < (TPB >> 5)) ? scr[l] : 0.0f;
  r = warpSum32(r);
  __syncthreads();
  return r;
}

// ---------------- kernel 1: per-(row,chunk) max & sum-exp ----------------
// grid = ROWS*NCHUNK blocks; stages 32KB of logits into LDS via the CDNA5
// async global->LDS path, then two passes out of LDS (1x HBM traffic).
__global__ void __launch_bounds__(TPB)
sal_partial_kernel(const float* __restrict__ logits, float2* __restrict__ partials) {
  __shared__ __align__(16) float sdata[CHUNK];
  __shared__ float scr[8];

  const int row   = blockIdx.x >> 2;
  const int chunk = blockIdx.x & 3;
  const float* gbase = logits + (size_t)row * VOCAB + (size_t)chunk * CHUNK;

  // Async copy: one b128 per lane per iteration. LDS byte address is the low
  // 32 bits of the generic pointer (generic LDS addr = {aperture, offset}).
  for (int i = threadIdx.x; i < CHUNK_V4; i += TPB) {
    unsigned loff = (unsigned)(uintptr_t)(const void*)(&sdata[i * 4]);
    const float* gp = gbase + (size_t)i * 4;
    asm volatile("global_load_async_to_lds_b128 %0, %1, off"
                 :: "v"(loff), "v"(gp) : "memory");
  }
  asm volatile("s_wait_asynccnt 0" ::: "memory");
  __syncthreads();  // make all waves' async LDS writes visible

  const float4* s4 = (const float4*)sdata;

  // pass 1: chunk max
  float m = -__builtin_inff();
  for (int i = threadIdx.x; i < CHUNK_V4; i += TPB) {
    float4 x = s4[i];
    m = fmaxf(m, fmaxf(fmaxf(x.x, x.y), fmaxf(x.z, x.w)));
  }
  m = blockMax(m, scr);

  // pass 2: chunk sum of exp(x - m)   (one v_exp_f32 per element)
  float ssum = 0.0f;
  for (int i = threadIdx.x; i < CHUNK_V4; i += TPB) {
    float4 x = s4[i];
    ssum += __expf(x.x - m) + __expf(x.y - m) + __expf(x.z - m) + __expf(x.w - m);
  }
  ssum = blockSum(ssum, scr);

  if (threadIdx.x == 0) partials[blockIdx.x] = make_float2(m, ssum);
}

// ---------------- kernel 2: per-row logZ + the 3 probabilities ----------------
__global__ void __launch_bounds__(TPB)
sal_row_kernel(const float* __restrict__ logits, const int* __restrict__ targets,
               const float2* __restrict__ partials,
               float* __restrict__ nll, float* __restrict__ pnl,
               float* __restrict__ peos) {
  const int row = blockIdx.x * TPB + threadIdx.x;
  if (row >= ROWS) return;

  float2 p[NCHUNK];
  float m = -__builtin_inff();
#pragma unroll
  for (int c = 0; c < NCHUNK; ++c) {
    p[c] = partials[row * NCHUNK + c];
    m = fmaxf(m, p[c].x);
  }
  float ssum = 0.0f;
#pragma unroll
  for (int c = 0; c < NCHUNK; ++c) ssum += p[c].y * __expf(p[c].x - m);
  const float logZ = m + __logf(ssum);

  const size_t base = (size_t)row * VOCAB;
  const int t = targets[row];
  const float lt   = logits[base + (size_t)t];
  const float lnl  = logits[base + NEWLINE_IDX];
  const float leos = logits[base + EOS_IDX];

  nll[row]  = logZ - lt;            // -log_softmax[target]
  pnl[row]  = __expf(lnl  - logZ);  // p_newline
  peos[row] = __expf(leos - logZ);  // p_eos
}

// ---------------- kernel 3: line-pos scan + all loss terms ----------------
__global__ void __launch_bounds__(TPB)
sal_final_kernel(const int* __restrict__ targets, const float* __restrict__ nll,
                 const float* __restrict__ pnl, const float* __restrict__ peos,
                 float* __restrict__ out) {
  __shared__ float linepos[ROWS];   // 16 KB
  __shared__ float scr[8];

  // sequential reset-on-newline counter, one thread per batch row (matches scan)
  if (threadIdx.x < BATCH) {
    const int b = threadIdx.x;
    float count = 0.0f;
    for (int s = 0; s < SEQ; ++s) {
      const int t = targets[b * SEQ + s];
      float pos;
      if (t == NEWLINE_IDX)      { pos = count; count = 0.0f; }
      else if (t == PAD_IDX)     { pos = 0.0f; }
      else                       { count += 1.0f; pos = count; }
      linepos[b * SEQ + s] = pos;
    }
  }
  __syncthreads();

  float ce_num = 0.0f, ce_den = 0.0f, lsum = 0.0f, lensum = 0.0f, msum = 0.0f;
  for (int i = threadIdx.x; i < ROWS; i += TPB) {
    const int t = targets[i];
    if (t == PAD_IDX) continue;     // valid==0 kills every term
    msum += 1.0f;

    // weighted CE
    const float w = (t == NEWLINE_IDX) ? 5.0f : 1.0f;
    ce_num += nll[i] * w;
    ce_den += w;

    // line-length loss
    const float lp = linepos[i];
    const float p  = pnl[i];
    const float late  = fmaxf(lp - 7.0f, 0.0f) * (1.0f - p) * 0.5f;      // / sigma=2
    const float ep    = fmaxf(2.0f - lp, 0.0f);                          // == shortness
    const float early = (__expf(ep) - 1.0f) * p * 5.0f;
    const float hard  = (lp < 2.0f) ? p * 100.0f * (1.0f + ep) : 0.0f;
    lsum += late + early + hard;

    // song-length loss (position = sequence index)
    const float pos      = (float)(i & (SEQ - 1));
    const float pressure = 1.0f / (1.0f + __expf(-(pos - 200.0f) * 0.05f));
    const float late_pen = pressure * (1.0f - peos[i]);
    const float early_eos =
        (pos < 50.0f) ? peos[i] * 100.0f * __expf((50.0f - pos) * 0.1f) : 0.0f;
    lensum += late_pen + early_eos;
  }

  ce_num = blockSum(ce_num, scr);
  ce_den = blockSum(ce_den, scr);
  lsum   = blockSum(lsum,   scr);
  lensum = blockSum(lensum, scr);
  msum   = blockSum(msum,   scr);

  if (threadIdx.x == 0) {
    const float inv = 1.0f / (msum + 1e-8f);
    out[0] = ce_num / ce_den + 0.1f * lsum * inv + 0.1f * lensum * inv;
  }
}

// ---------------- launch ----------------
extern "C" void kernel_launch(void* const* d_in, const int* in_sizes, int n_in,
                              void* d_out, int out_size, void* d_ws, size_t ws_size,
                              hipStream_t stream) {
  const float* logits  = (const float*)d_in[0];
  const int*   targets = (const int*)d_in[1];
  float*       out     = (float*)d_out;

  // workspace layout: [float2 partials: ROWS*NCHUNK] [nll][pnl][peos] (ROWS each)
  float2* partials = (float2*)d_ws;
  float*  nll  = (float*)((char*)d_ws + (size_t)ROWS * NCHUNK * sizeof(float2));
  float*  pnl  = nll + ROWS;
  float*  peos = pnl + ROWS;

  sal_partial_kernel<<<ROWS * NCHUNK, TPB, 0, stream>>>(logits, partials);
  sal_row_kernel<<<(ROWS + TPB - 1) / TPB, TPB, 0, stream>>>(logits, targets,
                                                             partials, nll, pnl, peos);
  sal_final_kernel<<<1, TPB, 0, stream>>>(targets, nll, pnl, peos, out);
}